// BoundaryOperator_36756330119900
// MI455X (gfx1250) — compile-verified
//
#include <hip/hip_runtime.h>
#include <stdint.h>

// ---------------------------------------------------------------------------
// SpMM scatter-add (COO segment-sum), D_FEAT = 64, f32.
//   out[r, :] += val * feat[c, :]  for each nonzero (r, c, val)
//
// Memory-bound (0.25 FLOP/byte): optimized for the MI455X memory path:
//  - GLOBAL_LOAD_ASYNC_TO_LDS_B32 double-buffered staging of index streams
//  - global_prefetch_b8 of upcoming randomly-gathered feature rows
//  - native no-return global_atomic_add_f32 scatter (dest fits in 192MB L2)
//  - wave32: one wave per nonzero, float2 per lane = one 256B row per wave
// ---------------------------------------------------------------------------

#define TILE   256   // nonzeros staged per block per iteration
#define BLOCK  256   // 8 waves (wave32)
#define NWAVES 8
#define PFDIST 4     // prefetch distance in wave-iterations

// Async DMA: global -> LDS, 4 bytes per lane. GV addressing mode (64-bit vaddr).
// Inline asm (not the clang builtin) so it works on both ROCm-7.2 and the
// amdgpu-toolchain lane. Tracked by ASYNCcnt.
__device__ __forceinline__ void async_load_b32(const void* g, void* lds) {
  asm volatile("global_load_async_to_lds_b32 %0, %1, off"
               :
               : "v"((unsigned)(uintptr_t)lds),          // LDS byte address
                 "v"((unsigned long long)(uintptr_t)g)   // 64-bit global addr
               : "memory");
}

// Wait until at most N async loads remain outstanding (per-wave counter).
__device__ __forceinline__ void wait_asynccnt_3() {
  asm volatile("s_wait_asynccnt 3" ::: "memory");
}
__device__ __forceinline__ void wait_asynccnt_0() {
  asm volatile("s_wait_asynccnt 0" ::: "memory");
}

__global__ __launch_bounds__(BLOCK) void boundary_spmm_scatter(
    const float* __restrict__ feat,   // [NUM_E, 64]
    const float* __restrict__ vals,   // [NNZ]
    const int*   __restrict__ rows,   // [NNZ]
    const int*   __restrict__ cols,   // [NNZ]
    float*       __restrict__ out,    // [NUM_V, 64], pre-zeroed
    int nnz, int numTiles) {
  __shared__ int   sRow[2][TILE];
  __shared__ int   sCol[2][TILE];
  __shared__ float sVal[2][TILE];

  const int t    = threadIdx.x;
  const int lane = t & 31;
  const int wave = t >> 5;

  int tile = blockIdx.x;
  if (tile >= numTiles) return;  // uniform per block

  // Prologue: stage first tile into buffer 0.
  // Index clamped (never predicated off) so every wave issues exactly 3 async
  // loads per tile -> ASYNCcnt is uniform and the waits below are exact.
  {
    const int gi  = tile * TILE + t;
    const int gic = gi < nnz ? gi : nnz - 1;
    async_load_b32(rows + gic, &sRow[0][t]);
    async_load_b32(cols + gic, &sCol[0][t]);
    async_load_b32(vals + gic, &sVal[0][t]);
  }

  int buf = 0;
  for (; tile < numTiles; tile += gridDim.x) {
    const int nextTile = tile + (int)gridDim.x;
    if (nextTile < numTiles) {           // uniform per block
      // Kick off next tile's DMA into the other buffer, then wait only for
      // the current tile's 3 loads (in-order completion).
      const int gi  = nextTile * TILE + t;
      const int gic = gi < nnz ? gi : nnz - 1;
      const int nb  = buf ^ 1;
      async_load_b32(rows + gic, &sRow[nb][t]);
      async_load_b32(cols + gic, &sCol[nb][t]);
      async_load_b32(vals + gic, &sVal[nb][t]);
      wait_asynccnt_3();
    } else {
      wait_asynccnt_0();
    }
    __syncthreads();  // all waves' staging of current buffer complete

    // One wave per nonzero: lane l owns features [2l, 2l+1] of the 64-wide row.
    const int base = tile * TILE;
    for (int j = wave; j < TILE; j += NWAVES) {
      const int gi = base + j;
      if (gi >= nnz) break;  // wave-uniform

      // Prefetch the feature row we'll gather PFDIST iterations from now
      // (its col index is already in LDS) -> global_prefetch_b8 into L2.
      const int jp = j + NWAVES * PFDIST;
      if (jp < TILE && base + jp < nnz) {
        const int colp = sCol[buf][jp];
        __builtin_prefetch(feat + (size_t)colp * 64 + lane * 2, 0, 1);
      }

      const int   col = sCol[buf][j];   // LDS broadcast (uniform addr in wave)
      const int   row = sRow[buf][j];
      const float val = sVal[buf][j];

      const float2 f = *(const float2*)(feat + (size_t)col * 64 + lane * 2);

      // No-return agent-scope f32 atomics; destination (128 MB) lives in L2.
      float* o = out + (size_t)row * 64 + lane * 2;
      __hip_atomic_fetch_add(o,     val * f.x, __ATOMIC_RELAXED,
                             __HIP_MEMORY_SCOPE_AGENT);
      __hip_atomic_fetch_add(o + 1, val * f.y, __ATOMIC_RELAXED,
                             __HIP_MEMORY_SCOPE_AGENT);
    }
    __syncthreads();  // done reading this buffer before it is re-filled
    buf ^= 1;
  }
}

extern "C" void kernel_launch(void* const* d_in, const int* in_sizes, int n_in,
                              void* d_out, int out_size, void* d_ws, size_t ws_size,
                              hipStream_t stream) {
  // setup_inputs() order: simplex_features, boundary_values, boundary_rows,
  //                       boundary_cols, num_out
  const float* feat = (const float*)d_in[0];
  const float* vals = (const float*)d_in[1];
  const int*   rows = (const int*)d_in[2];
  const int*   cols = (const int*)d_in[3];
  float*       out  = (float*)d_out;

  const int nnz = in_sizes[1];  // == in_sizes[2] == in_sizes[3]

  // d_out is poisoned by the harness; zero it (capturable async memset).
  hipMemsetAsync(d_out, 0, (size_t)out_size * sizeof(float), stream);

  const int numTiles = (nnz + TILE - 1) / TILE;
  int blocks = numTiles < 1536 ? numTiles : 1536;  // persistent grid-stride

  boundary_spmm_scatter<<<blocks, BLOCK, 0, stream>>>(
      feat, vals, rows, cols, out, nnz, numTiles);
}